// CARAFE3d_32899449487702
// MI455X (gfx1250) — compile-verified
//
#include <hip/hip_runtime.h>
#include <hip/hip_bf16.h>

typedef float v2f __attribute__((ext_vector_type(2)));
typedef float v8f __attribute__((ext_vector_type(8)));

// D = A(16x4 f32) * B(4x16 f32) + C(16x16 f32)
__device__ __forceinline__ v8f wmma_f32(v2f a, v2f b, v8f c) {
  return __builtin_amdgcn_wmma_f32_16x16x4_f32(false, a, false, b, (short)0, c,
                                               false, false);
}

#define SPATIAL 32768  // 32*32*32
#define MPAD 256       // conv output channels padded 216 -> 256 (16 M-tiles)
#define NWT (27 * 64 * MPAD)

// ---------------------------------------------------------------------------
// Prep: transpose/pad conv weights into WMMA-friendly layout + pad bias.
//   Wt[((koff*32 + cm/2)*MPAD + o)*2 + (cm&1)] = We[o][cm][koff]  (0 if o>=216)
//   bep[o] = be[o] (0 if o>=216)
// ---------------------------------------------------------------------------
__global__ __launch_bounds__(256) void prep_weights_kernel(
    const float* __restrict__ We, const float* __restrict__ be,
    float* __restrict__ Wt, float* __restrict__ bep) {
  int idx = blockIdx.x * 256 + threadIdx.x;
  if (idx < NWT) {
    int o = idx & (MPAD - 1);
    int cm = (idx >> 8) & 63;
    int koff = idx >> 14;  // 0..26
    float v = (o < 216) ? We[(size_t)o * 1728 + cm * 27 + koff] : 0.0f;
    Wt[((size_t)(koff * 32 + (cm >> 1)) * MPAD + o) * 2 + (cm & 1)] = v;
  } else if (idx < NWT + MPAD) {
    int o = idx - NWT;
    bep[o] = (o < 216) ? be[o] : 0.0f;
  }
}

// ---------------------------------------------------------------------------
// Generic small-M GEMM:  C[M][N] = A[M][K=64] * B[K][N] (+ bias[M])
// Block = 128 threads (4 waves), 64 N-cols. lB interleaved by k-parity.
// ---------------------------------------------------------------------------
template <int MTILES, bool BIAS>
__global__ __launch_bounds__(128) void gemm_f32_wmma_kernel(
    const float* __restrict__ A, const float* __restrict__ B,
    const float* __restrict__ bias, float* __restrict__ C, int N) {
  constexpr int M = MTILES * 16;
  __shared__ float lA[64 * 64];
  __shared__ float lB[64 * 64];  // [(k/2)*64 + col][k&1] pairs

  const int tid = threadIdx.x;
  const int lane = tid & 31;
  const int wave = tid >> 5;
  const int n0 = blockIdx.x * 64;
  const int ml = lane & 15;
  const int lhi = lane >> 4;  // 0/1

  for (int i = tid; i < M * 64; i += 128) lA[i] = A[i];
  for (int i = tid; i < 64 * 64; i += 128) {
    int k = i >> 6, col = i & 63;
    lB[((k >> 1) * 64 + col) * 2 + (k & 1)] = B[(size_t)k * N + n0 + col];
  }
  __syncthreads();

  int mt[MTILES], nt[MTILES];
#pragma unroll
  for (int i = 0; i < MTILES; ++i) {
    int p = wave + 4 * i;
    mt[i] = p % MTILES;
    nt[i] = p / MTILES;
  }

  v8f acc[MTILES];
#pragma unroll
  for (int i = 0; i < MTILES; ++i) {
    v8f c;
#pragma unroll
    for (int v = 0; v < 8; ++v)
      c[v] = BIAS ? bias[mt[i] * 16 + v + 8 * lhi] : 0.0f;
    acc[i] = c;
  }

#pragma unroll
  for (int k0 = 0; k0 < 64; k0 += 4) {
#pragma unroll
    for (int i = 0; i < MTILES; ++i) {
      v2f a = *(const v2f*)&lA[(mt[i] * 16 + ml) * 64 + k0 + 2 * lhi];
      v2f b = *(const v2f*)&lB[(((k0 >> 1) + lhi) * 64 + nt[i] * 16 + ml) * 2];
      acc[i] = wmma_f32(a, b, acc[i]);
    }
  }

#pragma unroll
  for (int i = 0; i < MTILES; ++i)
#pragma unroll
    for (int v = 0; v < 8; ++v) {
      int m = mt[i] * 16 + v + 8 * lhi;
      C[(size_t)m * N + n0 + nt[i] * 16 + ml] = acc[i][v];
    }
}

// ---------------------------------------------------------------------------
// 3x3x3 conv as im2col GEMM:  y2p[256][S] = Wt * im2col(y1) + bep
// One block (4 waves) per (d,h) row: 32 w-positions (2 N-tiles), 16 M-tiles,
// wave w owns tiles {w, w+4, w+8, w+12} — fully uniform, no guards.
// Per (kd,kh): zero-filled 64x34 LDS strip (k-parity interleaved) covers all
// three kw taps; A fragments are coalesced v2f loads from padded Wt.
// ---------------------------------------------------------------------------
__global__ __launch_bounds__(128) void conv3_wmma_kernel(
    const float* __restrict__ y1, const float* __restrict__ Wt,
    const float* __restrict__ bep, float* __restrict__ y2p) {
  __shared__ float strip[64 * 34];  // [(cm/2)*34 + pos][cm&1]

  const int tid = threadIdx.x;
  const int lane = tid & 31;
  const int wave = tid >> 5;
  const int d = blockIdx.x >> 5;
  const int h = blockIdx.x & 31;
  const int ml = lane & 15;
  const int lhi = lane >> 4;  // 0/1

  v8f acc[4][2];
#pragma unroll
  for (int t = 0; t < 4; ++t) {
    int ob = (wave + 4 * t) * 16;
#pragma unroll
    for (int j = 0; j < 2; ++j) {
      v8f c;
#pragma unroll
      for (int v = 0; v < 8; ++v) c[v] = bep[ob + v + 8 * lhi];
      acc[t][j] = c;
    }
  }

  for (int q = 0; q < 9; ++q) {
    const int kd = q / 3, kh = q % 3;
    const int dd = d + kd - 1, hh = h + kh - 1;
    const bool rowok = (dd >= 0 && dd < 32 && hh >= 0 && hh < 32);
    const int ddc = min(max(dd, 0), 31), hhc = min(max(hh, 0), 31);
    const float* src = y1 + ((size_t)ddc * 32 + hhc) * 32;

    __syncthreads();  // previous strip fully consumed
    for (int i = tid; i < 64 * 34; i += 128) {
      int cm = i / 34, pos = i % 34;
      int wsrc = pos - 1;
      int wc = min(max(wsrc, 0), 31);
      float ld = src[(size_t)cm * SPATIAL + wc];  // always-safe clamped load
      bool ok = rowok && (wsrc >= 0) && (wsrc < 32);
      strip[((cm >> 1) * 34 + pos) * 2 + (cm & 1)] = ok ? ld : 0.0f;
    }
    // prefetch next (kd,kh) row of y1 while this strip is consumed
    if (q < 8 && tid < 64) {
      int kd2 = (q + 1) / 3, kh2 = (q + 1) % 3;
      int d2 = min(max(d + kd2 - 1, 0), 31), h2 = min(max(h + kh2 - 1, 0), 31);
      __builtin_prefetch(y1 + ((size_t)tid * SPATIAL) + ((size_t)d2 * 32 + h2) * 32,
                         0, 0);
    }
    __syncthreads();

    const int koffb = kd * 9 + kh * 3;
#pragma unroll 2
    for (int k0 = 0; k0 < 64; k0 += 4) {
#pragma unroll
      for (int t = 0; t < 4; ++t) {
        int ob = (wave + 4 * t) * 16;
        // A-frag base: pair (cm = k0+2*lhi, k0+2*lhi+1), lane column ob+ml
        const float* ab =
            Wt + ((size_t)((k0 >> 1) + lhi) * MPAD + ob + ml) * 2 + koffb * (size_t)(32 * MPAD * 2);
#pragma unroll
        for (int kw = 0; kw < 3; ++kw) {
          v2f a = *(const v2f*)(ab + (size_t)kw * (32 * MPAD * 2));
#pragma unroll
          for (int j = 0; j < 2; ++j) {
            v2f b = *(const v2f*)&strip[(((k0 >> 1) + lhi) * 34 + j * 16 + ml + kw) * 2];
            acc[t][j] = wmma_f32(a, b, acc[t][j]);
          }
        }
      }
    }
  }

  const size_t sbase = ((size_t)d * 32 + h) * 32;
#pragma unroll
  for (int t = 0; t < 4; ++t) {
    int ob = (wave + 4 * t) * 16;
#pragma unroll
    for (int j = 0; j < 2; ++j)
#pragma unroll
      for (int v = 0; v < 8; ++v) {
        int o = ob + v + 8 * lhi;
        y2p[(size_t)o * SPATIAL + sbase + j * 16 + ml] = acc[t][j][v];
      }
  }
}

// ---------------------------------------------------------------------------
// Pixel-shuffle + softmax over 27 taps.
// y2p[n*8+sub][s]  ->  wsft[(s*27+n)*8 + sub]   (sub = rd*4+rh*2+rw)
// ---------------------------------------------------------------------------
__global__ __launch_bounds__(256) void softmax27_kernel(
    const float* __restrict__ y2p, float* __restrict__ wsft) {
  int t = blockIdx.x * 256 + threadIdx.x;  // 262144 total
  int sub = t >> 15;
  int s = t & (SPATIAL - 1);
  float v[27];
  float mx = -3.0e38f;
#pragma unroll
  for (int n = 0; n < 27; ++n) {
    v[n] = y2p[(size_t)(n * 8 + sub) * SPATIAL + s];
    mx = fmaxf(mx, v[n]);
  }
  float sum = 0.0f;
#pragma unroll
  for (int n = 0; n < 27; ++n) {
    v[n] = __expf(v[n] - mx);
    sum += v[n];
  }
  float inv = 1.0f / sum;
#pragma unroll
  for (int n = 0; n < 27; ++n)
    wsft[((size_t)s * 27 + n) * 8 + sub] = v[n] * inv;
}

// ---------------------------------------------------------------------------
// Final gather: out[o][2d+rd][2h+rh][2w+rw] = sum_n wsft[n]*Y[o][shift_n(d,h,w)]
// One 64-thread block per base cell.
// ---------------------------------------------------------------------------
__global__ __launch_bounds__(64) void carafe_apply_kernel(
    const float* __restrict__ Y, const float* __restrict__ wsft,
    float* __restrict__ out) {
  __shared__ float wl[216];      // [n*8+sub]
  __shared__ float yl[32 * 27];  // [o*27+n]
  int s = blockIdx.x;
  int d = s >> 10, h = (s >> 5) & 31, w = s & 31;
  int tid = threadIdx.x;

  for (int i = tid; i < 216; i += 64) wl[i] = wsft[(size_t)s * 216 + i];
  for (int i = tid; i < 32 * 27; i += 64) {
    int o = i / 27, n = i % 27;
    int nd = min(max(d + n / 9 - 1, 0), 31);
    int nh = min(max(h + (n / 3) % 3 - 1, 0), 31);
    int nw = min(max(w + n % 3 - 1, 0), 31);
    yl[i] = Y[(size_t)o * SPATIAL + ((size_t)nd * 32 + nh) * 32 + nw];
  }
  __syncthreads();

#pragma unroll
  for (int r = 0; r < 4; ++r) {
    int oi = tid + r * 64;  // 0..255
    int o = oi >> 3, sub = oi & 7;
    float acc = 0.0f;
#pragma unroll
    for (int n = 0; n < 27; ++n) acc += yl[o * 27 + n] * wl[n * 8 + sub];
    int rd = sub >> 2, rh = (sub >> 1) & 1, rw = sub & 1;
    size_t vox = ((size_t)(2 * d + rd) * 64 + (2 * h + rh)) * 64 + (2 * w + rw);
    out[(size_t)o * 262144 + vox] = acc;
  }
}

// ---------------------------------------------------------------------------
extern "C" void kernel_launch(void* const* d_in, const int* in_sizes, int n_in,
                              void* d_out, int out_size, void* d_ws,
                              size_t ws_size, hipStream_t stream) {
  (void)in_sizes; (void)n_in; (void)out_size; (void)ws_size;
  const float* x  = (const float*)d_in[0];  // (64, 32^3)
  const float* Wc = (const float*)d_in[1];  // (64, 64)
  const float* bc = (const float*)d_in[2];  // (64)
  const float* We = (const float*)d_in[3];  // (216, 64, 3,3,3)
  const float* be = (const float*)d_in[4];  // (216)
  const float* Wp = (const float*)d_in[5];  // (32, 64)
  float* out = (float*)d_out;               // (32, 64^3)

  float* ws = (float*)d_ws;
  float* y1   = ws;                               // 64  * 32768
  float* y2p  = y1 + (size_t)64 * SPATIAL;        // 256 * 32768 (padded)
  float* wsft = y2p + (size_t)MPAD * SPATIAL;     // 32768 * 216
  float* Yp   = wsft + (size_t)SPATIAL * 216;     // 32  * 32768
  float* Wt   = Yp + (size_t)32 * SPATIAL;        // 27*64*256
  float* bep  = Wt + (size_t)NWT;                 // 256

  // 0) pad+transpose conv weights/bias into WMMA layout
  prep_weights_kernel<<<(NWT + MPAD + 255) / 256, 256, 0, stream>>>(We, be, Wt, bep);
  // 1) y1 = Wc @ x + bc           (WMMA GEMM, M=64)
  gemm_f32_wmma_kernel<4, true><<<512, 128, 0, stream>>>(Wc, x, bc, y1, SPATIAL);
  // 2) y2p = conv3x3x3(y1)        (WMMA im2col GEMM, M=256 padded)
  conv3_wmma_kernel<<<1024, 128, 0, stream>>>(y1, Wt, bep, y2p);
  // 3) softmax + pixel shuffle
  softmax27_kernel<<<1024, 256, 0, stream>>>(y2p, wsft);
  // 4) Y = Wp @ x                 (WMMA GEMM, M=32) — projection commuted
  gemm_f32_wmma_kernel<2, false><<<512, 128, 0, stream>>>(Wp, x, nullptr, Yp, SPATIAL);
  // 5) out = sum_n wsft_n * shift_n(Y)
  carafe_apply_kernel<<<32768, 64, 0, stream>>>(Yp, wsft, out);
}